// down_block11_57458072486028
// MI455X (gfx1250) — compile-verified
//
#include <hip/hip_runtime.h>

// ---------------------------------------------------------------------------
// down_block11: PointNet++-style block for MI455X (gfx1250, wave32, WMMA)
// All GEMMs use V_WMMA_F32_16X16X4_F32 (fp32 matrix pipe).
// Each wave computes a 16x64 output strip (4 column tiles) so the A fragment
// is reused by 4 WMMAs per K-step.
// ---------------------------------------------------------------------------

typedef __attribute__((ext_vector_type(2))) float v2f;
typedef __attribute__((ext_vector_type(8))) float v8f;

#define NB    8
#define CIN   128
#define NPT   2048
#define KNN   8
#define SPT   512          // NPT / 4
#define CMID  131
#define BN    16384        // NB * NPT
#define COLSK 131072       // NB * NPT * KNN

// ---------------------------------------------------------------------------
// 1) Farthest point sampling: one block per batch, sequential 512 iterations.
//    pts[b][n][c] = feature flat reinterpreted (reshape, not transpose).
// ---------------------------------------------------------------------------
__global__ void fps_kernel(const float* __restrict__ feature, int* __restrict__ far_idx) {
  __shared__ float cen[CIN];
  __shared__ float dist[NPT];
  __shared__ float rmax[512];
  __shared__ int   ridx[512];
  __shared__ int   s_far;
  const int b = blockIdx.x;
  const int t = threadIdx.x;                       // 0..511
  const float* pts = feature + (long)b * NPT * CIN;
  for (int n = t; n < NPT; n += 512) dist[n] = 1e10f;
  if (t == 0) s_far = 0;
  __syncthreads();
  for (int it = 0; it < SPT; ++it) {
    const int far = s_far;
    if (t == 0) far_idx[b * SPT + it] = far;
    if (t < CIN) cen[t] = pts[(long)far * CIN + t];
    __syncthreads();
    float bestv = -1.0f;
    int   besti = 0;
    for (int n = t; n < NPT; n += 512) {
      const float* p = pts + (long)n * CIN;
      float d = 0.0f;
      for (int c = 0; c < CIN; c += 4) {
        float4 pv = *(const float4*)(p + c);
        float dx = pv.x - cen[c + 0];
        float dy = pv.y - cen[c + 1];
        float dz = pv.z - cen[c + 2];
        float dw = pv.w - cen[c + 3];
        d += dx * dx + dy * dy + dz * dz + dw * dw;
      }
      float nd = fminf(dist[n], d);
      dist[n] = nd;
      if (nd > bestv) { bestv = nd; besti = n; }   // n increasing -> ties keep first
    }
    rmax[t] = bestv; ridx[t] = besti;
    __syncthreads();
    for (int s = 256; s > 0; s >>= 1) {
      if (t < s) {
        float v2 = rmax[t + s]; int i2 = ridx[t + s];
        if (v2 > rmax[t] || (v2 == rmax[t] && i2 < ridx[t])) { rmax[t] = v2; ridx[t] = i2; }
      }
      __syncthreads();
    }
    if (t == 0) s_far = ridx[0];
    __syncthreads();
  }
}

// ---------------------------------------------------------------------------
// 2) Gather down_feats (B,S,128) and cand_xyz (B,S,3)
// ---------------------------------------------------------------------------
__global__ void gather_kernel(const float* __restrict__ feature,
                              const float* __restrict__ input,
                              const int* __restrict__ far_idx,
                              float* __restrict__ down_feats,
                              float* __restrict__ cand_xyz) {
  int i = blockIdx.x * 256 + threadIdx.x;          // b*512+s, < 4096
  if (i >= NB * SPT) return;
  int b = i >> 9;
  int far = far_idx[i];
  const float* src = feature + (long)b * (NPT * CIN) + (long)far * CIN;
  float* dst = down_feats + (long)i * CIN;
  for (int c = 0; c < CIN; ++c) dst[c] = src[c];
  for (int j = 0; j < 3; ++j)
    cand_xyz[i * 3 + j] = input[b * 3 * NPT + j * NPT + far];
}

// ---------------------------------------------------------------------------
// 3) kNN: for each (b,n), top-8 nearest of 512 candidates in 3D.
// ---------------------------------------------------------------------------
__global__ void knn_kernel(const float* __restrict__ input,
                           const float* __restrict__ cand_xyz,
                           int* __restrict__ knn_idx) {
  __shared__ float cx[SPT], cy[SPT], cz[SPT];
  const int b = blockIdx.y;
  const int n = blockIdx.x * blockDim.x + threadIdx.x;     // 0..2047
  for (int s = threadIdx.x; s < SPT; s += blockDim.x) {
    cx[s] = cand_xyz[(b * SPT + s) * 3 + 0];
    cy[s] = cand_xyz[(b * SPT + s) * 3 + 1];
    cz[s] = cand_xyz[(b * SPT + s) * 3 + 2];
  }
  __syncthreads();
  float px = input[b * 3 * NPT + 0 * NPT + n];
  float py = input[b * 3 * NPT + 1 * NPT + n];
  float pz = input[b * 3 * NPT + 2 * NPT + n];
  float bd[KNN]; int bi[KNN];
#pragma unroll
  for (int j = 0; j < KNN; ++j) { bd[j] = 1e30f; bi[j] = 0; }
  for (int s = 0; s < SPT; ++s) {
    float dx = px - cx[s], dy = py - cy[s], dz = pz - cz[s];
    float nd = dx * dx + dy * dy + dz * dz;
    int ni = s;
#pragma unroll
    for (int j = 0; j < KNN; ++j) {
      if (nd < bd[j]) {
        float td = bd[j]; int ti = bi[j];
        bd[j] = nd; bi[j] = ni;
        nd = td; ni = ti;
      }
    }
  }
#pragma unroll
  for (int j = 0; j < KNN; ++j)
    knn_idx[((long)b * NPT + n) * KNN + j] = bi[j];
}

// ---------------------------------------------------------------------------
// 4) Build feats (131 x COLSK), col = (b*N+n)*8+k
// ---------------------------------------------------------------------------
__global__ void feats_build_kernel(const float* __restrict__ down_feats,
                                   const float* __restrict__ cand_xyz,
                                   const float* __restrict__ input,
                                   const int* __restrict__ knn_idx,
                                   float* __restrict__ feats) {
  int col = blockIdx.x * 256 + threadIdx.x;        // < 131072
  if (col >= COLSK) return;
  int b = col >> 14;
  int rem = col & 16383;
  int n = rem >> 3;
  int s = knn_idx[col];
  const float* df = down_feats + ((long)(b * SPT + s)) * CIN;
  for (int c = 0; c < CIN; ++c)
    feats[(long)c * COLSK + col] = df[c];
  const float* cxp = cand_xyz + ((long)(b * SPT + s)) * 3;
  for (int j = 0; j < 3; ++j)
    feats[(long)(CIN + j) * COLSK + col] = cxp[j] - input[b * 3 * NPT + j * NPT + n];
}

// ---------------------------------------------------------------------------
// 5) skip_in = max over K (131 x BN)
// ---------------------------------------------------------------------------
__global__ void skip_max_kernel(const float* __restrict__ feats, float* __restrict__ skip_in) {
  int i = blockIdx.x * 256 + threadIdx.x;
  if (i >= CMID * BN) return;
  int c = i / BN;
  int bn = i % BN;
  const float* p = feats + (long)c * COLSK + (long)bn * KNN;
  float m = p[0];
#pragma unroll
  for (int k = 1; k < KNN; ++k) m = fmaxf(m, p[k]);
  skip_in[(long)c * BN + bn] = m;
}

// ---------------------------------------------------------------------------
// Generic fp32 WMMA GEMM: D = epilogue(A @ B).
// One wave computes a 16x64 strip (4 16x16 tiles, A fragment reused 4x).
// K-loop steps of 4 using V_WMMA_F32_16X16X4_F32, fp32 lane layout:
//   lanes 0-15: K = {k, k+1}, lanes 16-31: K = {k+2, k+3}.
// Branchless main loops (K rounded down to a multiple of 4; partial row
// tiles clamp the A row and zero via select); one guarded tail step covers
// Kd % 4. b_mode==1 implements B[r][c] = Bm[(r&255)*COLSK + c*8 + (r>>8)]
// (einsum('okw,bwnk->bon') view, Kd==2048, full row tiles) with purely
// incremental addressing.
// Epilogue: (+bias[r]) -> (*gamma[r]+beta[r]) -> relu -> (+add[r][c]).
// ---------------------------------------------------------------------------
__global__ void wmma_gemm_f32(
    const float* __restrict__ A, long ars, long acs, long abs_,
    const float* __restrict__ Bm, long brs, long bbs,
    float* __restrict__ D, long drs, long dbs,
    int M, int Kd, int ncols,
    const float* __restrict__ bias,
    const float* __restrict__ gamma, const float* __restrict__ beta,
    const float* __restrict__ addsrc, long addrs, long addbs,
    int do_relu, int b_mode) {
  const int lane = threadIdx.x & 31;
  const int wv   = threadIdx.x >> 5;
  const int col0 = (blockIdx.x * 4 + wv) * 64;     // 64-wide strip per wave
  if (col0 >= ncols) return;
  const int row0 = blockIdx.y * 16;
  const int bz   = blockIdx.z;
  const int half = lane >> 4;                       // 0: K={k,k+1}, 1: K={k+2,k+3}
  const int lid  = lane & 15;

  const float* Ab = A  + (long)bz * abs_;
  const float* Bb = Bm + (long)bz * bbs;

  v8f acc[4];
#pragma unroll
  for (int j = 0; j < 4; ++j) acc[j] = (v8f){0.f, 0.f, 0.f, 0.f, 0.f, 0.f, 0.f, 0.f};

  const int  arow    = row0 + lid;
  const bool arow_ok = (arow < M);
  const long arow_c  = arow_ok ? (long)arow : (long)(M - 1);  // clamped, in-bounds
  const int  bcol    = col0 + lid;                 // ncols is multiple of 64

  if (b_mode == 1) {
    // Kd == 2048, M multiple of 16 -> no guards at all.
    const float* aptr = Ab + arow_c * ars + (2 * half);
    for (int kk = 0; kk < 8; ++kk) {
      const float* bptr = Bb + (long)(2 * half) * COLSK + (long)bcol * 8 + kk;
#pragma unroll 2
      for (int w0 = 0; w0 < 256; w0 += 4) {
        v2f av; av.x = aptr[0];     av.y = aptr[1];
#pragma unroll
        for (int j = 0; j < 4; ++j) {
          v2f bv; bv.x = bptr[j * 128];   bv.y = bptr[j * 128 + COLSK];
          acc[j] = __builtin_amdgcn_wmma_f32_16x16x4_f32(false, av, false, bv,
                                                         (short)0, acc[j], false, false);
        }
        aptr += 4;
        bptr += 4 * (long)COLSK;
      }
    }
  } else {
    const int Kd_main = Kd & ~3;
    const float* aptr = Ab + arow_c * ars + (long)(2 * half) * acs;
    const float* bptr = Bb + (long)(2 * half) * brs + bcol;
    const long astep = 4 * acs;
    const long bstep = 4 * brs;
    if (row0 + 16 <= M) {
      // full row tile: completely unguarded main loop
#pragma unroll 2
      for (int k0 = 0; k0 < Kd_main; k0 += 4) {
        v2f av; av.x = aptr[0];   av.y = aptr[acs];
#pragma unroll
        for (int j = 0; j < 4; ++j) {
          v2f bv; bv.x = bptr[16 * j];   bv.y = bptr[brs + 16 * j];
          acc[j] = __builtin_amdgcn_wmma_f32_16x16x4_f32(false, av, false, bv,
                                                         (short)0, acc[j], false, false);
        }
        aptr += astep;
        bptr += bstep;
      }
    } else {
      // partial row tile: unconditional (clamped) loads + select-zero for A
#pragma unroll 2
      for (int k0 = 0; k0 < Kd_main; k0 += 4) {
        float a0 = aptr[0], a1 = aptr[acs];
        v2f av; av.x = arow_ok ? a0 : 0.f;
        av.y = arow_ok ? a1 : 0.f;
#pragma unroll
        for (int j = 0; j < 4; ++j) {
          v2f bv; bv.x = bptr[16 * j];   bv.y = bptr[brs + 16 * j];
          acc[j] = __builtin_amdgcn_wmma_f32_16x16x4_f32(false, av, false, bv,
                                                         (short)0, acc[j], false, false);
        }
        aptr += astep;
        bptr += bstep;
      }
    }
    if (Kd_main < Kd) {
      // single guarded tail step (Kd % 4 != 0, e.g. Kd = 131)
      const int ka = Kd_main + 2 * half;
      const bool k0ok = (ka < Kd);
      const bool k1ok = (ka + 1 < Kd);
      const long kc0 = k0ok ? (long)ka : (long)(Kd - 1);
      const long kc1 = k1ok ? (long)(ka + 1) : (long)(Kd - 1);
      float a0 = Ab[arow_c * ars + kc0 * acs];
      float a1 = Ab[arow_c * ars + kc1 * acs];
      v2f av; av.x = (arow_ok && k0ok) ? a0 : 0.f;
      av.y = (arow_ok && k1ok) ? a1 : 0.f;
#pragma unroll
      for (int j = 0; j < 4; ++j) {
        float b0 = Bb[kc0 * brs + bcol + 16 * j];
        float b1 = Bb[kc1 * brs + bcol + 16 * j];
        v2f bv; bv.x = k0ok ? b0 : 0.f;
        bv.y = k1ok ? b1 : 0.f;
        acc[j] = __builtin_amdgcn_wmma_f32_16x16x4_f32(false, av, false, bv,
                                                       (short)0, acc[j], false, false);
      }
    }
  }

  float* Db = D + (long)bz * dbs;
  const float* addb = addsrc ? (addsrc + (long)bz * addbs) : nullptr;
#pragma unroll
  for (int j = 0; j < 4; ++j) {
    const int bc = bcol + 16 * j;
#pragma unroll
    for (int v = 0; v < 8; ++v) {
      int r = row0 + v + 8 * half;
      if (r >= M) continue;
      float val = acc[j][v];
      if (bias)  val += bias[r];
      if (gamma) val = val * gamma[r] + beta[r];
      if (do_relu) val = fmaxf(val, 0.f);
      if (addb)  val += addb[(long)r * addrs + bc];
      Db[(long)r * drs + bc] = val;
    }
  }
}

// ---------------------------------------------------------------------------
// glob rows 128..130 = input (B,3,N) transposed into (c, b*N+n)
// ---------------------------------------------------------------------------
__global__ void glob_tail_kernel(const float* __restrict__ input, float* __restrict__ glob) {
  int i = blockIdx.x * 256 + threadIdx.x;          // < 3*BN
  if (i >= 3 * BN) return;
  int j = i / BN;
  int bn = i % BN;
  int b = bn >> 11, n = bn & 2047;
  glob[(long)(CIN + j) * BN + bn] = input[b * 3 * NPT + j * NPT + n];
}

// ---------------------------------------------------------------------------
// Row softmax over att[b][n][0..2047], in place (subtract max, like jax).
// ---------------------------------------------------------------------------
__global__ void softmax_rows_kernel(float* __restrict__ att) {
  __shared__ float red[256];
  const long row = blockIdx.x;                     // 0..BN-1
  float* p = att + row * NPT;
  const int t = threadIdx.x;
  float m = -1e30f;
  for (int i = t; i < NPT; i += 256) m = fmaxf(m, p[i]);
  red[t] = m; __syncthreads();
  for (int s = 128; s > 0; s >>= 1) { if (t < s) red[t] = fmaxf(red[t], red[t + s]); __syncthreads(); }
  m = red[0]; __syncthreads();
  float sum = 0.f;
  for (int i = t; i < NPT; i += 256) { float e = __expf(p[i] - m); p[i] = e; sum += e; }
  red[t] = sum; __syncthreads();
  for (int s = 128; s > 0; s >>= 1) { if (t < s) red[t] += red[t + s]; __syncthreads(); }
  const float inv = 1.0f / red[0];
  for (int i = t; i < NPT; i += 256) p[i] *= inv;
}

__global__ void col_sum_kernel(const float* __restrict__ att, float* __restrict__ colsum) {
  const int b = blockIdx.y;
  const int mcol = blockIdx.x * 256 + threadIdx.x;
  const float* p = att + (long)b * NPT * NPT + mcol;
  float s = 0.f;
  for (int r = 0; r < NPT; ++r) s += p[(long)r * NPT];
  colsum[b * NPT + mcol] = s;
}

__global__ void att_norm_kernel(float* __restrict__ att, const float* __restrict__ colsum) {
  long i = (long)blockIdx.x * 256 + threadIdx.x;   // < NB*NPT*NPT
  int mcol = (int)(i & (NPT - 1));
  int b = (int)(i >> 22);                          // NPT*NPT = 2^22
  att[i] /= (1e-9f + colsum[b * NPT + mcol]);
}

__global__ void diff_kernel(const float* __restrict__ glob, float* __restrict__ xr) {
  long i = (long)blockIdx.x * 256 + threadIdx.x;
  if (i >= (long)CMID * BN) return;
  xr[i] = glob[i] - xr[i];
}

// ---------------------------------------------------------------------------
// host: kernel_launch
// ---------------------------------------------------------------------------
extern "C" void kernel_launch(void* const* d_in, const int* in_sizes, int n_in,
                              void* d_out, int out_size, void* d_ws, size_t ws_size,
                              hipStream_t stream) {
  const float* feature = (const float*)d_in[0];
  const float* input   = (const float*)d_in[1];
  // d_in[2] rel_pos_emb is unused by the reference
  const float* w_skip  = (const float*)d_in[3];
  const float* g_skip  = (const float*)d_in[4];
  const float* b_skip  = (const float*)d_in[5];
  const float* w_m1    = (const float*)d_in[6];
  const float* w_m2    = (const float*)d_in[7];
  const float* g_m2    = (const float*)d_in[8];
  const float* b_m2    = (const float*)d_in[9];
  const float* w_out1  = (const float*)d_in[10];
  const float* g_out1  = (const float*)d_in[11];
  const float* b_out1  = (const float*)d_in[12];
  const float* w_out2  = (const float*)d_in[13];
  const float* g_out2  = (const float*)d_in[14];
  const float* b_out2  = (const float*)d_in[15];
  const float* w_q     = (const float*)d_in[16];
  const float* w_k     = (const float*)d_in[17];
  const float* w_v     = (const float*)d_in[18];
  const float* b_v     = (const float*)d_in[19];
  const float* w_t     = (const float*)d_in[20];
  const float* b_t     = (const float*)d_in[21];
  const float* g_ra    = (const float*)d_in[22];
  const float* b_ra    = (const float*)d_in[23];
  const float* w_mlp1  = (const float*)d_in[24];
  const float* w_conv2 = (const float*)d_in[25];
  const float* g_conv2 = (const float*)d_in[26];
  const float* b_conv2 = (const float*)d_in[27];
  float* out = (float*)d_out;

  // --- carve workspace ---
  char* wsp = (char*)d_ws;
  auto carve = [&](size_t bytes) -> void* {
    void* p = (void*)wsp;
    wsp += (bytes + 255) & ~(size_t)255;
    return p;
  };
  int*   far_idx    = (int*)  carve((size_t)NB * SPT * 4);
  float* down_feats = (float*)carve((size_t)NB * SPT * CIN * 4);
  float* cand_xyz   = (float*)carve((size_t)NB * SPT * 3 * 4);
  int*   knn_idx    = (int*)  carve((size_t)COLSK * 4);
  float* feats      = (float*)carve((size_t)CMID * COLSK * 4);
  float* skip_in    = (float*)carve((size_t)CMID * BN * 4);
  float* skipbuf    = (float*)carve((size_t)256 * BN * 4);
  float* f1buf      = (float*)carve((size_t)256 * COLSK * 4);
  float* fbuf       = (float*)carve((size_t)256 * COLSK * 4);
  float* g1buf      = (float*)carve((size_t)256 * BN * 4);
  float* glob       = (float*)carve((size_t)CMID * BN * 4);
  float* Qbuf       = (float*)carve((size_t)32 * BN * 4);
  float* Kbuf       = (float*)carve((size_t)32 * BN * 4);
  float* Vbuf       = (float*)carve((size_t)CMID * BN * 4);
  float* att        = (float*)carve((size_t)NB * NPT * NPT * 4);
  float* colsum     = (float*)carve((size_t)NB * NPT * 4);
  float* xr         = (float*)carve((size_t)CMID * BN * 4);
  float* ra         = (float*)carve((size_t)CMID * BN * 4);
  float* out1       = (float*)carve((size_t)256 * BN * 4);

  auto gemm = [&](const float* A, long ars, long acs, long abs_,
                  const float* Bm, long brs, long bbs,
                  float* Dp, long drs, long dbs,
                  int M, int Kd, int nc, int nbatch,
                  const float* bias, const float* gam, const float* bet,
                  const float* add, long addrs, long addbs, int relu, int bmode) {
    dim3 grid((nc + 255) / 256, (M + 15) / 16, nbatch);
    wmma_gemm_f32<<<grid, 128, 0, stream>>>(A, ars, acs, abs_, Bm, brs, bbs,
                                            Dp, drs, dbs, M, Kd, nc,
                                            bias, gam, bet, add, addrs, addbs, relu, bmode);
  };

  // 1) FPS
  fps_kernel<<<NB, 512, 0, stream>>>(feature, far_idx);
  // 2) gather
  gather_kernel<<<(NB * SPT + 255) / 256, 256, 0, stream>>>(feature, input, far_idx,
                                                            down_feats, cand_xyz);
  // 3) kNN
  knn_kernel<<<dim3(NPT / 256, NB), 256, 0, stream>>>(input, cand_xyz, knn_idx);
  // 4) feats
  feats_build_kernel<<<COLSK / 256, 256, 0, stream>>>(down_feats, cand_xyz, input,
                                                      knn_idx, feats);
  // 5) skip_in = max_k
  skip_max_kernel<<<(CMID * BN + 255) / 256, 256, 0, stream>>>(feats, skip_in);

  // 6) skip = relu(bn(w_skip @ skip_in))            (256 x BN)
  gemm(w_skip, CMID, 1, 0, skip_in, BN, 0, skipbuf, BN, 0,
       256, CMID, BN, 1, nullptr, g_skip, b_skip, nullptr, 0, 0, 1, 0);
  // 7) f1 = relu(w_m1 @ feats)                      (256 x COLSK)
  gemm(w_m1, CMID, 1, 0, feats, COLSK, 0, f1buf, COLSK, 0,
       256, CMID, COLSK, 1, nullptr, nullptr, nullptr, nullptr, 0, 0, 1, 0);
  // 8) f = relu(bn(w_m2 @ f1))                      (256 x COLSK)
  gemm(w_m2, 256, 1, 0, f1buf, COLSK, 0, fbuf, COLSK, 0,
       256, 256, COLSK, 1, nullptr, g_m2, b_m2, nullptr, 0, 0, 1, 0);
  // 9) g1 = relu(bn(einsum(w_out1, f))) + skip      (256 x BN), b_mode=1
  gemm(w_out1, 2048, 1, 0, fbuf, 0, 0, g1buf, BN, 0,
       256, 2048, BN, 1, nullptr, g_out1, b_out1, skipbuf, BN, 0, 1, 1);
  // 10) glob[0:128] = relu(bn(w_out2 @ g1))         (128 x BN)
  gemm(w_out2, 256, 1, 0, g1buf, BN, 0, glob, BN, 0,
       128, 256, BN, 1, nullptr, g_out2, b_out2, nullptr, 0, 0, 1, 0);
  // 11) glob[128:131] = input
  glob_tail_kernel<<<(3 * BN + 255) / 256, 256, 0, stream>>>(input, glob);
  // 12) Q = w_q @ glob                              (32 x BN)
  gemm(w_q, CMID, 1, 0, glob, BN, 0, Qbuf, BN, 0,
       32, CMID, BN, 1, nullptr, nullptr, nullptr, nullptr, 0, 0, 0, 0);
  // 13) K = w_k @ glob                              (32 x BN)
  gemm(w_k, CMID, 1, 0, glob, BN, 0, Kbuf, BN, 0,
       32, CMID, BN, 1, nullptr, nullptr, nullptr, nullptr, 0, 0, 0, 0);
  // 14) V = w_v @ glob + b_v                        (131 x BN)
  gemm(w_v, CMID, 1, 0, glob, BN, 0, Vbuf, BN, 0,
       CMID, CMID, BN, 1, b_v, nullptr, nullptr, nullptr, 0, 0, 0, 0);
  // 15) S[b] = Q_b^T @ K_b                          (2048 x 2048 per batch)
  gemm(Qbuf, 1, BN, NPT, Kbuf, BN, NPT, att, NPT, (long)NPT * NPT,
       NPT, 32, NPT, NB, nullptr, nullptr, nullptr, nullptr, 0, 0, 0, 0);
  // 16) row softmax
  softmax_rows_kernel<<<BN, 256, 0, stream>>>(att);
  // 17) column sums, 18) PCT normalization
  col_sum_kernel<<<dim3(NPT / 256, NB), 256, 0, stream>>>(att, colsum);
  att_norm_kernel<<<(int)(((long)NB * NPT * NPT) / 256), 256, 0, stream>>>(att, colsum);
  // 19) x_r[b] = V_b @ att_b                        (131 x 2048 per batch)
  gemm(Vbuf, BN, 1, NPT, att, NPT, (long)NPT * NPT, xr, BN, NPT,
       CMID, NPT, NPT, NB, nullptr, nullptr, nullptr, nullptr, 0, 0, 0, 0);
  // 20) xr = glob - xr
  diff_kernel<<<(CMID * BN + 255) / 256, 256, 0, stream>>>(glob, xr);
  // 21) ra = glob + relu(bn(w_t @ xr + b_t))        (131 x BN)
  gemm(w_t, CMID, 1, 0, xr, BN, 0, ra, BN, 0,
       CMID, CMID, BN, 1, b_t, g_ra, b_ra, glob, BN, 0, 1, 0);
  // 22) out1 = relu(w_mlp1 @ ra)                    (256 x BN)
  gemm(w_mlp1, CMID, 1, 0, ra, BN, 0, out1, BN, 0,
       256, CMID, BN, 1, nullptr, nullptr, nullptr, nullptr, 0, 0, 1, 0);
  // 23) out = relu(bn(w_conv2 @ out1)) -> d_out (B,128,N), per-batch columns
  gemm(w_conv2, 256, 1, 0, out1, BN, NPT, out, NPT, (long)128 * NPT,
       128, 256, NPT, NB, nullptr, g_conv2, b_conv2, nullptr, 0, 0, 1, 0);

  (void)in_sizes; (void)n_in; (void)out_size; (void)ws_size;
}